// PMA_9337258901602
// MI455X (gfx1250) — compile-verified
//
#include <hip/hip_runtime.h>
#include <math.h>

// ---------------------------------------------------------------------------
// PMA (GMM-seeded multihead attention pooling) for MI455X / gfx1250.
// fp32 WMMA (v_wmma_f32_16x16x4_f32) for both fused GEMM passes,
// async global->LDS staging, 128-row tiles to minimize weight re-reads.
// ---------------------------------------------------------------------------

typedef __attribute__((ext_vector_type(2))) float v2f;
typedef __attribute__((ext_vector_type(8))) float v8f;

#define N_INST   2048
#define SEEDS    4
#define DIM      512
#define NHEAD    8
#define HEADDIM  64
#define BATCH    16
#define NROWS    (BATCH * N_INST * SEEDS)      // 131072 global rows of X
#define ROWS_PB  128                           // rows per block (8 WMMA tiles)
#define NTILES   (ROWS_PB / 16)
#define LDS_STR  516                           // 4-bank skew, 16B-aligned rows
#define LOG2PI   1.8378770664093453f
#define BN_EPS   1e-5f
#define LL_SCALE 0.044194173824159216f         // 1/sqrt(512)

__device__ __forceinline__ v8f wmma4(v2f a, v2f b, v8f c) {
  // D = A(16x4,f32) x B(4x16,f32) + C(16x16,f32)
  return __builtin_amdgcn_wmma_f32_16x16x4_f32(false, a, false, b,
                                               (short)0, c, false, false);
}

// Async-copy 128 rows x 512 floats of X into LDS (row stride LDS_STR).
// Uses CDNA5 GLOBAL_LOAD_ASYNC_TO_LDS_B128 (ASYNCcnt), no VGPR round-trip.
__device__ __forceinline__ void stage_rows_async(
    float (*xs)[LDS_STR], const float* __restrict__ src, int tid) {
#pragma unroll 1
  for (int i = tid; i < ROWS_PB * (DIM / 4); i += 256) {
    const int r = i >> 7, c4 = i & 127;
    const unsigned lds_off = (unsigned)(size_t)&xs[r][c4 * 4];  // low 32b = LDS offset
    const float* g = src + (size_t)r * DIM + c4 * 4;
    asm volatile("global_load_async_to_lds_b128 %0, %1, off"
                 :: "v"(lds_off), "v"(g) : "memory");
  }
  asm volatile("s_wait_asynccnt 0x0" ::: "memory");
}

// 128 rows (LDS) x 64 cols (head slice of W) x 512 k, fp32 WMMA.
// W is [DIM][DIM] row-major; output col c uses W[c][k] (i.e. X @ W.T).
// One set of B fragments feeds 8 row tiles -> 32 WMMAs per 4 b64 W-loads.
__device__ __forceinline__ void gemm_tiles(
    const float (*xs)[LDS_STR], const float* __restrict__ W,
    int colbase, int lane16, int khalf, v8f acc[NTILES][4]) {
  const float* wp0 = W + (size_t)(colbase +  0 + lane16) * DIM + khalf;
  const float* wp1 = W + (size_t)(colbase + 16 + lane16) * DIM + khalf;
  const float* wp2 = W + (size_t)(colbase + 32 + lane16) * DIM + khalf;
  const float* wp3 = W + (size_t)(colbase + 48 + lane16) * DIM + khalf;
#pragma unroll 2
  for (int k0 = 0; k0 < DIM; k0 += 4) {
    v2f b0 = *(const v2f*)(wp0 + k0);
    v2f b1 = *(const v2f*)(wp1 + k0);
    v2f b2 = *(const v2f*)(wp2 + k0);
    v2f b3 = *(const v2f*)(wp3 + k0);
    v2f a[NTILES];
#pragma unroll
    for (int t = 0; t < NTILES; ++t)
      a[t] = *(const v2f*)&xs[t * 16 + lane16][k0 + khalf];
#pragma unroll
    for (int t = 0; t < NTILES; ++t) {
      acc[t][0] = wmma4(a[t], b0, acc[t][0]);
      acc[t][1] = wmma4(a[t], b1, acc[t][1]);
      acc[t][2] = wmma4(a[t], b2, acc[t][2]);
      acc[t][3] = wmma4(a[t], b3, acc[t][3]);
    }
  }
}

// ---------------------------------------------------------------------------
// prep: sigma = softplus(S); qhalf = 0.5/sigma^2; mu_eff = M - bk (folds K bias)
//       c1[s*8+h] = log_softmax(p)[s] + sum_d(-0.5*log2pi - log sigma)
// ---------------------------------------------------------------------------
__global__ __launch_bounds__(256) void k_prep(
    const float* __restrict__ M, const float* __restrict__ Sp,
    const float* __restrict__ p, const float* __restrict__ bk,
    float* __restrict__ mu_eff, float* __restrict__ qhalf,
    float* __restrict__ c1) {
  __shared__ float lgs[SEEDS * DIM];
  const int tid = threadIdx.x;
  for (int i = tid; i < SEEDS * DIM; i += 256) {
    int d = i & (DIM - 1);
    float x   = Sp[i];
    float sig = fmaxf(x, 0.f) + log1pf(expf(-fabsf(x)));   // softplus
    mu_eff[i] = M[i] - bk[d];
    qhalf[i]  = 0.5f / (sig * sig);
    lgs[i]    = logf(sig);
  }
  __syncthreads();
  if (tid < SEEDS * NHEAD) {
    const int s = tid >> 3, h = tid & 7;
    float mx  = fmaxf(fmaxf(p[0], p[1]), fmaxf(p[2], p[3]));
    float den = expf(p[0] - mx) + expf(p[1] - mx) + expf(p[2] - mx) + expf(p[3] - mx);
    float acc = (p[s] - mx) - logf(den);                    // log softmax
    for (int d = 0; d < HEADDIM; ++d)
      acc += -0.5f * LOG2PI - lgs[s * DIM + h * HEADDIM + d];
    c1[tid] = acc;
  }
}

// ---------------------------------------------------------------------------
// Pass 1: K = X@Wk.T (bias folded into mu_eff), reduced to
//         ll[row*8+h] = c1[s,h] - sum_d qhalf*(K-mu_eff)^2.  128 rows/block.
// ---------------------------------------------------------------------------
__global__ __launch_bounds__(256)
__attribute__((amdgpu_waves_per_eu(1)))
void k_gemm_ll(
    const float* __restrict__ X, const float* __restrict__ Wk,
    const float* __restrict__ mu_eff, const float* __restrict__ qhalf,
    const float* __restrict__ c1, float* __restrict__ ll) {
  __shared__ __align__(16) float xs[ROWS_PB][LDS_STR];
  const int tid     = threadIdx.x;
  const int h       = tid >> 5;                // wave == head
  const int lane    = tid & 31;
  const int lane16  = lane & 15;
  const int khalf   = (lane < 16) ? 0 : 2;
  const int rbase   = blockIdx.x * ROWS_PB;
  const int colbase = h * HEADDIM;

  stage_rows_async(xs, X + (size_t)rbase * DIM, tid);
  __syncthreads();

  v8f acc[NTILES][4];
#pragma unroll
  for (int t = 0; t < NTILES; ++t)
#pragma unroll
    for (int s = 0; s < 4; ++s) acc[t][s] = {};

  gemm_tiles(xs, Wk, colbase, lane16, khalf, acc);

  // per-lane GMM constants for the 4 seeds at this lane's columns
  float muv[4][4], qv[4][4], c1h[4];
#pragma unroll
  for (int ss = 0; ss < 4; ++ss) {
    c1h[ss] = c1[ss * NHEAD + h];
#pragma unroll
    for (int sub = 0; sub < 4; ++sub) {
      int col = colbase + sub * 16 + lane16;
      muv[ss][sub] = mu_eff[ss * DIM + col];
      qv[ss][sub]  = qhalf[ss * DIM + col];
    }
  }

#pragma unroll
  for (int t = 0; t < NTILES; ++t) {
#pragma unroll
    for (int v = 0; v < 8; ++v) {
      const int ss = v & 3;                    // seed of this C-row (128-aligned base)
      float tmp = 0.f;
#pragma unroll
      for (int sub = 0; sub < 4; ++sub) {
        float d = acc[t][sub][v] - muv[ss][sub];
        tmp = fmaf(d * d, qv[ss][sub], tmp);
      }
      tmp += __shfl_xor(tmp, 1, 32);
      tmp += __shfl_xor(tmp, 2, 32);
      tmp += __shfl_xor(tmp, 4, 32);
      tmp += __shfl_xor(tmp, 8, 32);
      if (lane16 == 0) {                       // lane 0 -> row v, lane 16 -> row v+8
        int row = rbase + t * 16 + v + ((lane >= 16) ? 8 : 0);
        ll[(size_t)row * NHEAD + h] = c1h[ss] - tmp;
      }
    }
  }
}

// ---------------------------------------------------------------------------
// Softmax over n (2048) per (b,s,h); in-place ll -> A.
// ---------------------------------------------------------------------------
__global__ __launch_bounds__(256) void k_softmax(float* __restrict__ ll) {
  __shared__ float red[256];
  const int g = blockIdx.x;                    // 512 groups
  const int b = g >> 5, rem = g & 31, s = rem >> 3, h = rem & 7;
  const int tid = threadIdx.x;

  float vals[8];
  float m = -3.4e38f;
#pragma unroll
  for (int j = 0; j < 8; ++j) {
    int n = tid + j * 256;
    size_t idx = ((size_t)((b * N_INST + n) * SEEDS + s)) * NHEAD + h;
    float v = ll[idx] * LL_SCALE;
    vals[j] = v;
    m = fmaxf(m, v);
  }
  red[tid] = m; __syncthreads();
  for (int off = 128; off > 0; off >>= 1) {
    if (tid < off) red[tid] = fmaxf(red[tid], red[tid + off]);
    __syncthreads();
  }
  m = red[0]; __syncthreads();

  float sum = 0.f;
#pragma unroll
  for (int j = 0; j < 8; ++j) { vals[j] = expf(vals[j] - m); sum += vals[j]; }
  red[tid] = sum; __syncthreads();
  for (int off = 128; off > 0; off >>= 1) {
    if (tid < off) red[tid] += red[tid + off];
    __syncthreads();
  }
  const float inv = 1.f / red[0];
#pragma unroll
  for (int j = 0; j < 8; ++j) {
    int n = tid + j * 256;
    size_t idx = ((size_t)((b * N_INST + n) * SEEDS + s)) * NHEAD + h;
    ll[idx] = vals[j] * inv;
  }
}

// ---------------------------------------------------------------------------
// Pass 2: V = X@Wv.T fused with A-weighted accumulation over n.
// 128 rows/block, deterministic per-chunk partials.
// Opart layout: [b][chunk(64)][s(4)][DIM]
// ---------------------------------------------------------------------------
__global__ __launch_bounds__(256)
__attribute__((amdgpu_waves_per_eu(1)))
void k_weighted_v(
    const float* __restrict__ X, const float* __restrict__ Wv,
    const float* __restrict__ A, float* __restrict__ Opart) {
  __shared__ __align__(16) float xs[ROWS_PB][LDS_STR];
  const int tid     = threadIdx.x;
  const int h       = tid >> 5;
  const int lane    = tid & 31;
  const int lane16  = lane & 15;
  const int khalf   = (lane < 16) ? 0 : 2;
  const int rbase   = blockIdx.x * ROWS_PB;
  const int b       = rbase >> 13;             // / 8192 rows per batch
  const int chunk   = (rbase & 8191) >> 7;     // / 128
  const int colbase = h * HEADDIM;

  stage_rows_async(xs, X + (size_t)rbase * DIM, tid);
  __syncthreads();

  v8f acc[NTILES][4];
#pragma unroll
  for (int t = 0; t < NTILES; ++t)
#pragma unroll
    for (int s = 0; s < 4; ++s) acc[t][s] = {};

  gemm_tiles(xs, Wv, colbase, lane16, khalf, acc);

  float part[4][4];
#pragma unroll
  for (int ss = 0; ss < 4; ++ss)
#pragma unroll
    for (int sub = 0; sub < 4; ++sub) part[ss][sub] = 0.f;

#pragma unroll
  for (int t = 0; t < NTILES; ++t) {
#pragma unroll
    for (int v = 0; v < 8; ++v) {
      int row = rbase + t * 16 + v + ((lane >= 16) ? 8 : 0);
      float w = A[(size_t)row * NHEAD + h];
      const int ss = v & 3;
      part[ss][0] += w * acc[t][0][v];
      part[ss][1] += w * acc[t][1][v];
      part[ss][2] += w * acc[t][2][v];
      part[ss][3] += w * acc[t][3][v];
    }
  }

#pragma unroll
  for (int ss = 0; ss < 4; ++ss) {
#pragma unroll
    for (int sub = 0; sub < 4; ++sub) {
      float val = part[ss][sub] + __shfl_xor(part[ss][sub], 16, 32);
      if (lane < 16) {
        size_t o = ((size_t)(b * 64 + chunk) * 4 + ss) * DIM + colbase + sub * 16 + lane;
        Opart[o] = val;
      }
    }
  }
}

// ---------------------------------------------------------------------------
// O0[b][s*512+col] = mu[s,col] + bv[col] + sum_{chunk} Opart   (deterministic)
// ---------------------------------------------------------------------------
__global__ __launch_bounds__(512) void k_reduce_chunks(
    const float* __restrict__ Opart, const float* __restrict__ M,
    const float* __restrict__ bv, float* __restrict__ O0) {
  const int bs = blockIdx.x;                   // b*4+s
  const int b = bs >> 2, s = bs & 3;
  const int col = threadIdx.x;                 // 512
  float acc = M[s * DIM + col] + bv[col];      // sum_n A == 1 -> bias folds here
  for (int c = 0; c < 64; ++c)
    acc += Opart[((size_t)(b * 64 + c) * 4 + s) * DIM + col];
  O0[(size_t)bs * DIM + col] = acc;
}

// ---------------------------------------------------------------------------
// BatchNorm over batch (16) for 2048 features (training-mode, biased var).
// ---------------------------------------------------------------------------
__global__ __launch_bounds__(256) void k_bn(
    const float* __restrict__ in, const float* __restrict__ g,
    const float* __restrict__ bb, float* __restrict__ out) {
  const int f = blockIdx.x * 256 + threadIdx.x;       // 2048 features
  float x[BATCH], mean = 0.f;
#pragma unroll
  for (int b = 0; b < BATCH; ++b) { x[b] = in[b * 2048 + f]; mean += x[b]; }
  mean *= (1.f / BATCH);
  float var = 0.f;
#pragma unroll
  for (int b = 0; b < BATCH; ++b) { float d = x[b] - mean; var += d * d; }
  var *= (1.f / BATCH);
  const float inv = g[f] / sqrtf(var + BN_EPS);
  const float bias = bb[f];
#pragma unroll
  for (int b = 0; b < BATCH; ++b)
    out[b * 2048 + f] = (x[b] - mean) * inv + bias;
}

// ---------------------------------------------------------------------------
// O2 = O1 + relu(O1 @ Wo.T + bo)   (64 rows x 512 cols, tiny)
// ---------------------------------------------------------------------------
__global__ __launch_bounds__(256) void k_out_proj(
    const float* __restrict__ O1, const float* __restrict__ Wo,
    const float* __restrict__ bo, float* __restrict__ O2) {
  __shared__ float row[DIM];
  const int r = blockIdx.x;                    // b*4+s
  const int tid = threadIdx.x;
  row[tid] = O1[(size_t)r * DIM + tid];
  row[tid + 256] = O1[(size_t)r * DIM + tid + 256];
  __syncthreads();
#pragma unroll
  for (int cc = 0; cc < 2; ++cc) {
    const int c = tid + cc * 256;
    const float* w = Wo + (size_t)c * DIM;
    float acc = bo[c];
    for (int k = 0; k < DIM; ++k) acc = fmaf(row[k], w[k], acc);
    O2[(size_t)r * DIM + c] = row[c] + fmaxf(acc, 0.f);
  }
}

// ---------------------------------------------------------------------------
extern "C" void kernel_launch(void* const* d_in, const int* in_sizes, int n_in,
                              void* d_out, int out_size, void* d_ws, size_t ws_size,
                              hipStream_t stream) {
  const float* X  = (const float*)d_in[0];
  const float* M  = (const float*)d_in[1];
  const float* Sp = (const float*)d_in[2];
  const float* p  = (const float*)d_in[3];
  const float* Wk = (const float*)d_in[4];
  const float* bk = (const float*)d_in[5];
  const float* Wv = (const float*)d_in[6];
  const float* bv = (const float*)d_in[7];
  const float* Wo = (const float*)d_in[8];
  const float* bo = (const float*)d_in[9];
  const float* g0 = (const float*)d_in[10];
  const float* b0 = (const float*)d_in[11];
  const float* g1 = (const float*)d_in[12];
  const float* b1 = (const float*)d_in[13];

  float* ws     = (float*)d_ws;
  float* mu_eff = ws;                              // 2048
  float* qhalf  = ws + 2048;                       // 2048
  float* c1     = ws + 4096;                       // 32 (padded to 64)
  float* llA    = ws + 4160;                       // 131072*8 = 1,048,576
  float* Opart  = llA + (size_t)NROWS * NHEAD;     // 16*64*4*512 = 2,097,152
  float* O0     = Opart + (size_t)16 * 64 * 4 * DIM;   // 32768
  float* O1     = O0 + 64 * DIM;                   // 32768
  float* O2     = O1 + 64 * DIM;                   // 32768

  k_prep<<<1, 256, 0, stream>>>(M, Sp, p, bk, mu_eff, qhalf, c1);
  k_gemm_ll<<<NROWS / ROWS_PB, 256, 0, stream>>>(X, Wk, mu_eff, qhalf, c1, llA);
  k_softmax<<<BATCH * SEEDS * NHEAD, 256, 0, stream>>>(llA);
  k_weighted_v<<<NROWS / ROWS_PB, 256, 0, stream>>>(X, Wv, llA, Opart);
  k_reduce_chunks<<<64, 512, 0, stream>>>(Opart, M, bv, O0);
  k_bn<<<8, 256, 0, stream>>>(O0, g0, b0, O1);
  k_out_proj<<<64, 256, 0, stream>>>(O1, Wo, bo, O2);
  k_bn<<<8, 256, 0, stream>>>(O2, g1, b1, (float*)d_out);
}